// SeesawLossWithLogits_2078764171361
// MI455X (gfx1250) — compile-verified
//
#include <hip/hip_runtime.h>
#include <hip/hip_bf16.h>
#include <stdint.h>

// ---------------------------------------------------------------------------
// Seesaw loss, algebraically reduced: only denom[b, t_b] is ever needed, so
// the (B,C)x(C,C) GEMM collapses to one weighted dot product per row.
// Memory-bound: 14.6 MB of logits, staged row-by-row into LDS via the CDNA5
// Tensor Data Mover (tensor_load_to_lds + s_wait_tensorcnt).
// ---------------------------------------------------------------------------

typedef unsigned int v4u __attribute__((ext_vector_type(4)));
typedef int          v8i __attribute__((ext_vector_type(8)));
typedef int          v4i __attribute__((ext_vector_type(4)));

#define P_EXP 0.8f
#define EPS   1e-6f

// pw[c] = n_c ^ 0.8  (tiny precompute; removes all per-element pow)
__global__ void seesaw_pow_kernel(const int* __restrict__ counts,
                                  float* __restrict__ pw, int C) {
    int c = blockIdx.x * blockDim.x + threadIdx.x;
    if (c < C) pw[c] = __powf((float)counts[c], P_EXP);
}

__global__ __launch_bounds__(256)
void seesaw_row_kernel(const float* __restrict__ logits,
                       const int*   __restrict__ targets,
                       const int*   __restrict__ counts,
                       const float* __restrict__ pw,
                       float*       __restrict__ loss_out,
                       int C) {
    __shared__ float row[1792];   // 7168 B >= C*4 = 7136 B
    __shared__ float red[8];
    __shared__ float bcast;

    const int tid = threadIdx.x;
    const int b   = blockIdx.x;

    // ---- Stage logits row b into LDS with the Tensor Data Mover (wave 0) ----
    if (tid < 32) {
#if defined(__HIP_DEVICE_COMPILE__) && __has_builtin(__builtin_amdgcn_tensor_load_to_lds)
        uint32_t lds_off = (uint32_t)(uintptr_t)(void*)&row[0];   // addr[31:0] = LDS offset
        uint64_t ga = (uint64_t)(uintptr_t)(logits + (size_t)b * (size_t)C);

        v4u g0;
        g0.x = 1u;                                   // count=1 (valid), user descriptor
        g0.y = lds_off;                              // lds_addr [63:32]
        g0.z = (uint32_t)ga;                         // global_addr[31:0]
        g0.w = (uint32_t)((ga >> 32) & 0x1FFFFFFu)   // global_addr[56:32]
             | (2u << 30);                           // type = 2 ("image")

        v8i g1;
        g1[0] = (int)(2u << 16);                     // workgroup_mask=0, data_size=2 (4B)
        g1[1] = (int)(((uint32_t)C & 0xFFFFu) << 16);// tensor_dim0[15:0] @ bits[63:48]
        g1[2] = (int)(((uint32_t)C >> 16)            // tensor_dim0[31:16] @ bits[79:64]
             | (1u << 16));                          // tensor_dim1 = 1    @ bits[111:80]
        g1[3] = (int)(((uint32_t)C & 0xFFFFu) << 16);// tile_dim0 = C      @ bits[127:112]
        g1[4] = 1;                                   // tile_dim1 = 1, tile_dim2 = 0
        g1[5] = C;                                   // tensor_dim0_stride[31:0]
        g1[6] = 0;                                   // stride0 hi / stride1 lo
        g1[7] = 0;

        v4i g2 = {0, 0, 0, 0};
        v4i g3 = {0, 0, 0, 0};
        v8i g4 = {0, 0, 0, 0, 0, 0, 0, 0};           // extra group (clang-23 6-arg form)

        __builtin_amdgcn_tensor_load_to_lds(g0, g1, g2, g3, g4, 0);
        __builtin_amdgcn_s_wait_tensorcnt(0);        // TENSORcnt -> 0: tile is in LDS
#else
        for (int j = tid; j < C; j += 32) row[j] = logits[(size_t)b * C + j];
#endif
    }
    __syncthreads();

    // ---- Pass 1: row max (stable softmax shift) ----
    float m = -3.402823466e38f;
    for (int j = tid; j < C; j += 256) m = fmaxf(m, row[j]);
#pragma unroll
    for (int o = 16; o > 0; o >>= 1) m = fmaxf(m, __shfl_xor(m, o, 32));
    if ((tid & 31) == 0) red[tid >> 5] = m;
    __syncthreads();
    if (tid == 0) {
        float mm = red[0];
#pragma unroll
        for (int w = 1; w < 8; ++w) mm = fmaxf(mm, red[w]);
        bcast = mm;
    }
    __syncthreads();
    const float rowmax = bcast;

    // ---- Pass 2: denom[b] = sum_j s[t,j] * exp(l[b,j]-max) ----
    const int   t       = targets[b];
    const int   cnt_t   = counts[t];
    const float inv_pwt = 1.0f / pw[t];

    float acc = 0.0f;
    for (int j = tid; j < C; j += 256) {
        float e   = __expf(row[j] - rowmax);
        float sij = (cnt_t > counts[j]) ? pw[j] * inv_pwt : 1.0f;
        acc += sij * e;
    }
#pragma unroll
    for (int o = 16; o > 0; o >>= 1) acc += __shfl_xor(acc, o, 32);
    if ((tid & 31) == 0) red[tid >> 5] = acc;
    __syncthreads();

    if (tid == 0) {
        float denom = 0.0f;
#pragma unroll
        for (int w = 0; w < 8; ++w) denom += red[w];
        float et    = __expf(row[t] - rowmax);
        float sigma = et / (denom + EPS);
        loss_out[b] = -__logf(sigma + EPS);
    }
}

// Deterministic fixed-tree mean over B per-sample losses.
__global__ __launch_bounds__(256)
void seesaw_mean_kernel(const float* __restrict__ losses, float* __restrict__ out, int B) {
    __shared__ float red[8];
    const int tid = threadIdx.x;
    float a = 0.0f;
    for (int i = tid; i < B; i += 256) a += losses[i];
#pragma unroll
    for (int o = 16; o > 0; o >>= 1) a += __shfl_xor(a, o, 32);
    if ((tid & 31) == 0) red[tid >> 5] = a;
    __syncthreads();
    if (tid == 0) {
        float s = 0.0f;
#pragma unroll
        for (int w = 0; w < 8; ++w) s += red[w];
        out[0] = s / (float)B;
    }
}

extern "C" void kernel_launch(void* const* d_in, const int* in_sizes, int n_in,
                              void* d_out, int out_size, void* d_ws, size_t ws_size,
                              hipStream_t stream) {
    const float* logits  = (const float*)d_in[0];
    const int*   targets = (const int*)  d_in[1];
    const int*   counts  = (const int*)  d_in[2];

    const int B = in_sizes[1];   // 2048
    const int C = in_sizes[2];   // 1784

    float* losses = (float*)d_ws;      // B floats
    float* pw     = losses + B;        // C floats

    seesaw_pow_kernel <<<(C + 255) / 256, 256, 0, stream>>>(counts, pw, C);
    seesaw_row_kernel <<<B, 256, 0, stream>>>(logits, targets, counts, pw, losses, C);
    seesaw_mean_kernel<<<1, 256, 0, stream>>>(losses, (float*)d_out, B);
}